// GaussianAgg_4380866642195
// MI455X (gfx1250) — compile-verified
//
#include <hip/hip_runtime.h>
#include <stdint.h>

// ---------------- problem constants (from reference) ----------------
#define KCH   16          // K channels per pixel
#define KP1   17          // K + background
#define NSAMP 16          // NB_SAMPLES
#define TPB   256         // threads per block = pixels per tile
#define TILE_ELEMS (TPB * KCH)   // 4096 floats = 16 KB per input array
#define EPSV  1e-10f

#ifndef __has_builtin
#define __has_builtin(x) 0
#endif

#if __has_builtin(__builtin_amdgcn_tensor_load_to_lds)
#define USE_TDM 1
#else
#define USE_TDM 0
#endif

typedef unsigned int u32x4 __attribute__((ext_vector_type(4)));
typedef int          i32x4 __attribute__((ext_vector_type(4)));
typedef int          i32x8 __attribute__((ext_vector_type(8)));

// ---------------- TDM: DMA one 4096-float row (16 KB) global -> LDS --------
#if USE_TDM
__device__ __forceinline__ void tdm_load_tile(const void* gptr, uint32_t lds_byte_off) {
  uint64_t ga = (uint64_t)(uintptr_t)gptr;
  // D# group0: count=1, lds_addr, global_addr[56:0], type=2 ("image")
  u32x4 g0 = { 1u,
               lds_byte_off,
               (uint32_t)ga,
               (uint32_t)((ga >> 32) & 0x01FFFFFFu) | (2u << 30) };
  // D# group1: workgroup_mask=0, data_size=2 (4B), tensor_dim0=4096,
  // tensor_dim1=1, tile_dim0=4096, tile_dim1=1, tensor_dim0_stride=4096
  i32x8 g1 = { 0x00020000,
               (int)(TILE_ELEMS << 16),   // w1[31:16] = tensor_dim0 lo
               (int)(1u << 16),           // w2[31:16] = tensor_dim1 lo = 1
               (int)(TILE_ELEMS << 16),   // w3[31:16] = tile_dim0
               1,                         // w4[15:0]  = tile_dim1 = 1
               TILE_ELEMS,                // w5        = tensor_dim0_stride lo
               0, 0 };
  i32x4 gz = { 0, 0, 0, 0 };
#if __clang_major__ >= 23
  i32x8 gz8 = { 0, 0, 0, 0, 0, 0, 0, 0 };
  __builtin_amdgcn_tensor_load_to_lds(g0, g1, gz, gz, gz8, 0);
#else
  __builtin_amdgcn_tensor_load_to_lds(g0, g1, gz, gz, 0);
#endif
}

__device__ __forceinline__ void tdm_wait_all() {
#if __has_builtin(__builtin_amdgcn_s_wait_tensorcnt)
  __builtin_amdgcn_s_wait_tensorcnt(0);
#else
  asm volatile("s_wait_tensorcnt 0" ::: "memory");
#endif
}
#endif // USE_TDM

// ---------------- Philox4x32-7 counter-based RNG ---------------------------
__device__ __forceinline__ u32x4 philox7(uint32_t c0, uint32_t c1,
                                         uint32_t c2, uint32_t c3,
                                         uint32_t k0, uint32_t k1) {
#pragma unroll
  for (int r = 0; r < 7; ++r) {
    uint32_t h0 = __umulhi(0xD2511F53u, c0), l0 = 0xD2511F53u * c0;
    uint32_t h1 = __umulhi(0xCD9E8D57u, c2), l1 = 0xCD9E8D57u * c2;
    uint32_t n0 = h1 ^ c1 ^ k0;
    uint32_t n2 = h0 ^ c3 ^ k1;
    c0 = n0; c1 = l1; c2 = n2; c3 = l0;
    k0 += 0x9E3779B9u; k1 += 0xBB67AE85u;
  }
  u32x4 out = { c0, c1, c2, c3 };
  return out;
}

// Box-Muller: two u32 -> two N(0,1). v_sin/v_cos take input in revolutions,
// so the [0,1) uniform feeds them directly (no 2*pi, no range reduction).
__device__ __forceinline__ void boxmuller(uint32_t a, uint32_t b,
                                          float& n0, float& n1) {
  float u = (float)a * 0x1p-32f + 0x1p-33f;                 // (0,1)
  // -2*ln(u) = -2*ln2 * log2(u)
  float r = __builtin_amdgcn_sqrtf(-1.3862943611f * __builtin_amdgcn_logf(u));
  float t = (float)b * 0x1p-32f;                            // turns in [0,1)
  n0 = r * __builtin_amdgcn_cosf(t);
  n1 = r * __builtin_amdgcn_sinf(t);
}

// ---------------- main kernel ----------------------------------------------
__global__ __launch_bounds__(TPB)
void gauss_agg_kernel(const float* __restrict__ zbuf,
                      const float* __restrict__ prob,
                      const float* __restrict__ mask,
                      const float* __restrict__ gamma_p,
                      const float* __restrict__ alpha_p,
                      const float* __restrict__ zfar_p,
                      const float* __restrict__ znear_p,
                      float* __restrict__ out) {
  __shared__ __attribute__((aligned(16))) float shz[TILE_ELEMS];
  __shared__ __attribute__((aligned(16))) float shp[TILE_ELEMS];
  __shared__ __attribute__((aligned(16))) float shm[TILE_ELEMS];

  const int tid  = threadIdx.x;
  const int tile = blockIdx.x;
  const size_t base = (size_t)tile * TILE_ELEMS;

#if USE_TDM
  if (tid == 0) {
    tdm_load_tile(zbuf + base, (uint32_t)(uintptr_t)shz);
    tdm_load_tile(prob + base, (uint32_t)(uintptr_t)shp);
    tdm_load_tile(mask + base, (uint32_t)(uintptr_t)shm);
    tdm_wait_all();
  }
#else
  {
    const float4* gz4 = (const float4*)(zbuf + base);
    const float4* gp4 = (const float4*)(prob + base);
    const float4* gm4 = (const float4*)(mask + base);
    float4* sz4 = (float4*)shz;
    float4* sp4 = (float4*)shp;
    float4* sm4 = (float4*)shm;
    for (int i = tid; i < TILE_ELEMS / 4; i += TPB) {
      sz4[i] = gz4[i]; sp4[i] = gp4[i]; sm4[i] = gm4[i];
    }
  }
#endif
  __syncthreads();

  const float gamma = gamma_p[0];
  const float goa   = gamma / alpha_p[0];
  const float zfar  = zfar_p[0];
  const float inv_range = 1.0f / (zfar - znear_p[0]);

  // z_inv, z_inv_max (clipped to EPS), z_map + background channel
  float zm[KP1];
  float zmax = EPSV;
#pragma unroll
  for (int k = 0; k < KCH; ++k) {
    float zi = (zfar - shz[tid * KCH + k]) * inv_range * shm[tid * KCH + k];
    zm[k] = zi;
    zmax = fmaxf(zmax, zi);
  }
#pragma unroll
  for (int k = 0; k < KCH; ++k) {
    // ln(p) = ln2 * log2(p)
    float lp = 0.69314718056f * __builtin_amdgcn_logf(shp[tid * KCH + k]);
    zm[k] = goa * lp + zm[k] - zmax;
  }
  zm[KCH] = EPSV - zmax;  // background

  const uint32_t pix = (uint32_t)(tile * TPB + tid);

  float cnt[KP1];
#pragma unroll
  for (int k = 0; k < KP1; ++k) cnt[k] = 0.0f;

  for (int s = 0; s < NSAMP; ++s) {
    float v[20];
    // 5 Philox calls -> 20 normals (17 used)
#pragma unroll
    for (int j = 0; j < 5; ++j) {
      u32x4 r4 = philox7(pix, (uint32_t)s, (uint32_t)j, 0x1BADB002u,
                         0xDEADBEEFu, 0x0BADF00Du);
      float n0, n1, n2, n3;
      boxmuller(r4.x, r4.y, n0, n1);
      boxmuller(r4.z, r4.w, n2, n3);
      v[4 * j + 0] = n0; v[4 * j + 1] = n1;
      v[4 * j + 2] = n2; v[4 * j + 3] = n3;
    }
#pragma unroll
    for (int k = 0; k < KP1; ++k) v[k] = zm[k] + gamma * v[k];
    float vmax = v[0];
#pragma unroll
    for (int k = 1; k < KP1; ++k) vmax = fmaxf(vmax, v[k]);
#pragma unroll
    for (int k = 0; k < KP1; ++k) cnt[k] += (v[k] >= vmax) ? 1.0f : 0.0f;
  }

  float* op = out + (size_t)pix * KP1;
#pragma unroll
  for (int k = 0; k < KP1; ++k) op[k] = cnt[k] * (1.0f / NSAMP);
}

// ---------------- launcher -------------------------------------------------
extern "C" void kernel_launch(void* const* d_in, const int* in_sizes, int n_in,
                              void* d_out, int out_size, void* d_ws, size_t ws_size,
                              hipStream_t stream) {
  const float* zbuf  = (const float*)d_in[0];
  const float* prob  = (const float*)d_in[1];
  const float* mask  = (const float*)d_in[2];
  const float* gamma = (const float*)d_in[3];
  const float* alpha = (const float*)d_in[4];
  const float* zfar  = (const float*)d_in[5];
  const float* znear = (const float*)d_in[6];
  float* out = (float*)d_out;

  const int npix   = in_sizes[0] / KCH;   // 262144
  const int blocks = npix / TPB;          // 1024

  gauss_agg_kernel<<<dim3(blocks), dim3(TPB), 0, stream>>>(
      zbuf, prob, mask, gamma, alpha, zfar, znear, out);
}